// LePEAttention_62130996904236
// MI455X (gfx1250) — compile-verified
//
#include <hip/hip_runtime.h>
#include <hip/hip_bf16.h>

typedef __attribute__((ext_vector_type(2))) float v2f;
typedef __attribute__((ext_vector_type(8))) float v8f;

#define DIMC 128
#define RESHW 64
#define IMGPIX 4096              // 64*64
#define NWIN 256                 // 16 batches * 16 column windows
#define LWIN 256                 // H_SP*W_SP
#define HEADS 4
#define HDIM 32
#define IMG_ELEMS ((size_t)16 * DIMC * IMGPIX)   // 8,388,608

// ---------------------------------------------------------------------------
// Kernel 1: conv1x1  (q_map = wq@x + bq  or  v_map = wv@x + bv, per blockIdx.z)
// ---------------------------------------------------------------------------
__global__ void __launch_bounds__(256)
conv1x1_kernel(const float* __restrict__ x,
               const float* __restrict__ wq, const float* __restrict__ bq,
               const float* __restrict__ wv, const float* __restrict__ bv,
               float* __restrict__ q_map, float* __restrict__ v_map) {
    __shared__ float Ws[DIMC * DIMC];            // 64 KB
    const int tid = threadIdx.x;
    const float* W;  const float* bias;  float* dst;
    if (blockIdx.z == 0) { W = wq; bias = bq; dst = q_map; }
    else                 { W = wv; bias = bv; dst = v_map; }
    for (int i = tid; i < DIMC * DIMC; i += 256) Ws[i] = W[i];
    __syncthreads();

    const int gp  = blockIdx.x * 256 + tid;      // global spatial position
    const int b   = gp >> 12;
    const int pix = gp & (IMGPIX - 1);
    const float* xb = x   + (size_t)b * DIMC * IMGPIX + pix;
    float*       db = dst + (size_t)b * DIMC * IMGPIX + pix;

    for (int ot = 0; ot < 16; ++ot) {
        const int o0 = ot * 8;
        float acc[8];
#pragma unroll
        for (int j = 0; j < 8; ++j) acc[j] = bias[o0 + j];
        for (int c = 0; c < DIMC; ++c) {
            const float xv = xb[(size_t)c * IMGPIX];
#pragma unroll
            for (int j = 0; j < 8; ++j) acc[j] += Ws[(o0 + j) * DIMC + c] * xv;
        }
#pragma unroll
        for (int j = 0; j < 8; ++j) db[(size_t)(o0 + j) * IMGPIX] = acc[j];
    }
}

// ---------------------------------------------------------------------------
// Kernel 2: depthwise 5x5 (pad=2) on v_map -> lepe
// ---------------------------------------------------------------------------
__global__ void __launch_bounds__(256)
dw5x5_kernel(const float* __restrict__ v_map,
             const float* __restrict__ w_lepe, const float* __restrict__ b_lepe,
             float* __restrict__ lepe) {
    __shared__ float wsm[25];
    const int bc  = blockIdx.x;                  // b*128 + c
    const int c   = bc & (DIMC - 1);
    const int tid = threadIdx.x;
    if (tid < 25) wsm[tid] = w_lepe[c * 25 + tid];
    __syncthreads();
    const float  bias = b_lepe[c];
    const float* img  = v_map + (size_t)bc * IMGPIX;
    float*       out  = lepe  + (size_t)bc * IMGPIX;

    for (int k = 0; k < 16; ++k) {
        const int px = tid + k * 256;
        const int h = px >> 6, w = px & 63;
        float s = bias;
#pragma unroll
        for (int i = 0; i < 5; ++i) {
            const int hh = h + i - 2;
            if (hh < 0 || hh > 63) continue;
#pragma unroll
            for (int j = 0; j < 5; ++j) {
                const int ww = w + j - 2;
                if (ww < 0 || ww > 63) continue;
                s += wsm[i * 5 + j] * img[hh * 64 + ww];
            }
        }
        out[px] = s;
    }
}

// ---------------------------------------------------------------------------
// Kernel 3: fused windowed attention, one block per (window, head), 8 waves.
//   K = x+lepe (transposed in LDS), V = v_map, Q strip = (q_map+lepe)*scale.
//   scores via V_WMMA_F32_16X16X4_F32 ; fused softmax + temp*attn_input ;
//   out strip via WMMA ; scatter out+lepe through windows2img mapping.
// ---------------------------------------------------------------------------
__global__ void __launch_bounds__(256)
lepe_attn_kernel(const float* __restrict__ x,
                 const float* __restrict__ q_map,
                 const float* __restrict__ v_map,
                 const float* __restrict__ lepe,
                 const float* __restrict__ attn_in,
                 const float* __restrict__ temperature,
                 float* __restrict__ out_img,
                 float* __restrict__ out_attn) {
    __shared__ float Kt[HDIM * 260];     // K^T: [d][l], padded 256->260
    __shared__ float Vs[LWIN * 33];      // V:   [l][d], padded 32->33
    __shared__ float Ss[16 * 260];       // score/attn strip [row][l]
    __shared__ float Qs[16 * 36];        // Q strip [row][d], padded 32->36

    const int tid  = threadIdx.x;
    const int lane = tid & 31;
    const int wave = tid >> 5;
    const int win  = blockIdx.x;         // 0..255
    const int head = blockIdx.y;         // 0..3
    const int b    = win >> 4;
    const int w0   = (win & 15) * 4;     // first abs column of window
    const size_t img_base = (size_t)b * DIMC * IMGPIX;
    const int c0   = head * HDIM;
    const float temp  = temperature[head];
    const float scale = 0.17677669529663687f;   // 32^-0.5

    // ---- load K^T and V once per block (float4 per (channel,row)) ----
    for (int idx = tid; idx < HDIM * RESHW; idx += 256) {
        const int h     = idx & 63;
        const int c_off = idx >> 6;
        const size_t g  = img_base + (size_t)(c0 + c_off) * IMGPIX + h * 64 + w0;
        const float4 kx = *(const float4*)(x + g);
        const float4 kl = *(const float4*)(lepe + g);
        const float4 vv = *(const float4*)(v_map + g);
        const int l0 = h * 4;
        Kt[c_off * 260 + l0 + 0] = kx.x + kl.x;
        Kt[c_off * 260 + l0 + 1] = kx.y + kl.y;
        Kt[c_off * 260 + l0 + 2] = kx.z + kl.z;
        Kt[c_off * 260 + l0 + 3] = kx.w + kl.w;
        Vs[(l0 + 0) * 33 + c_off] = vv.x;
        Vs[(l0 + 1) * 33 + c_off] = vv.y;
        Vs[(l0 + 2) * 33 + c_off] = vv.z;
        Vs[(l0 + 3) * 33 + c_off] = vv.w;
    }
    __syncthreads();

    const int mrow = lane & 15;                  // M (or N) index within tile
    const int kk0  = (lane < 16) ? 0 : 2;        // f32 WMMA K sub-index per lane half

    for (int s = 0; s < 16; ++s) {
        // ---- load Q strip: (q_map + lepe) * scale ----
        for (int idx = tid; idx < 16 * HDIM; idx += 256) {
            const int row = idx >> 5, d = idx & 31;
            const int l = s * 16 + row;
            const size_t g = img_base + (size_t)(c0 + d) * IMGPIX + (l >> 2) * 64 + w0 + (l & 3);
            Qs[row * 36 + d] = (q_map[g] + lepe[g]) * scale;
        }
        __syncthreads();

        // ---- scores strip 16x256 : 16 tiles, 2 per wave, K=32 ----
#pragma unroll
        for (int t = 0; t < 2; ++t) {
            const int n0 = (wave * 2 + t) * 16;
            v8f acc = {};
#pragma unroll
            for (int k0 = 0; k0 < HDIM; k0 += 4) {
                v2f a, bb;
                a.x  = Qs[mrow * 36 + k0 + kk0];
                a.y  = Qs[mrow * 36 + k0 + kk0 + 1];
                bb.x = Kt[(k0 + kk0) * 260 + n0 + mrow];
                bb.y = Kt[(k0 + kk0 + 1) * 260 + n0 + mrow];
                acc = __builtin_amdgcn_wmma_f32_16x16x4_f32(
                        false, a, false, bb, (short)0, acc, false, false);
            }
#pragma unroll
            for (int r = 0; r < 8; ++r) {
                const int m = r + ((lane < 16) ? 0 : 8);
                Ss[m * 260 + n0 + mrow] = acc[r];
            }
        }
        __syncthreads();

        // ---- softmax over 256 cols + temperature*attn_input, stream attn out ----
        {
            const int row = tid >> 4;
            const int j   = tid & 15;
            float ex[16];
            float vmax = -3.4e38f;
#pragma unroll
            for (int t = 0; t < 16; ++t) {
                const float v = Ss[row * 260 + j + t * 16];
                ex[t] = v;
                vmax = fmaxf(vmax, v);
            }
#pragma unroll
            for (int off = 8; off > 0; off >>= 1)
                vmax = fmaxf(vmax, __shfl_xor(vmax, off, 16));
            float sum = 0.f;
#pragma unroll
            for (int t = 0; t < 16; ++t) { ex[t] = __expf(ex[t] - vmax); sum += ex[t]; }
#pragma unroll
            for (int off = 8; off > 0; off >>= 1)
                sum += __shfl_xor(sum, off, 16);
            const float inv = 1.0f / sum;
            const int l = s * 16 + row;
            const size_t rowbase = ((size_t)win * HEADS + head) * 65536 + (size_t)l * 256;
            const float* arow = attn_in + rowbase;
            float*       orow = out_attn + rowbase;
#pragma unroll
            for (int t = 0; t < 16; ++t) {
                const int col = j + t * 16;
                const float p = ex[t] * inv + temp * arow[col];
                Ss[row * 260 + col] = p;
                orow[col] = p;
            }
        }
        __syncthreads();

        // ---- out strip 16x32 = attn(16x256) @ V(256x32), waves 0..1 ----
        if (wave < 2) {
            const int n0 = wave * 16;
            v8f acc = {};
            for (int l0 = 0; l0 < LWIN; l0 += 4) {
                v2f a, bb;
                a.x  = Ss[mrow * 260 + l0 + kk0];
                a.y  = Ss[mrow * 260 + l0 + kk0 + 1];
                bb.x = Vs[(l0 + kk0) * 33 + n0 + mrow];
                bb.y = Vs[(l0 + kk0 + 1) * 33 + n0 + mrow];
                acc = __builtin_amdgcn_wmma_f32_16x16x4_f32(
                        false, a, false, bb, (short)0, acc, false, false);
            }
#pragma unroll
            for (int r = 0; r < 8; ++r) {
                const int m = r + ((lane < 16) ? 0 : 8);
                const int l = s * 16 + m;
                const int d = n0 + mrow;
                const size_t g = img_base + (size_t)(c0 + d) * IMGPIX + (l >> 2) * 64 + w0 + (l & 3);
                out_img[g] = acc[r] + lepe[g];
            }
        }
        __syncthreads();
    }
}

// ---------------------------------------------------------------------------
extern "C" void kernel_launch(void* const* d_in, const int* in_sizes, int n_in,
                              void* d_out, int out_size, void* d_ws, size_t ws_size,
                              hipStream_t stream) {
    const float* x           = (const float*)d_in[0];
    const float* attn_input  = (const float*)d_in[1];
    const float* wq          = (const float*)d_in[2];
    const float* bq          = (const float*)d_in[3];
    const float* wv          = (const float*)d_in[4];
    const float* bv          = (const float*)d_in[5];
    const float* w_lepe      = (const float*)d_in[6];
    const float* b_lepe      = (const float*)d_in[7];
    const float* temperature = (const float*)d_in[8];

    float* out_img  = (float*)d_out;
    float* out_attn = out_img + IMG_ELEMS;

    float* q_map = (float*)d_ws;
    float* v_map = q_map + IMG_ELEMS;
    float* lepe  = v_map + IMG_ELEMS;

    // 1) q_map / v_map : 65536 positions, 256 per block, z selects q vs v
    conv1x1_kernel<<<dim3(256, 1, 2), 256, 0, stream>>>(
        x, wq, bq, wv, bv, q_map, v_map);

    // 2) lepe = depthwise5x5(v_map)
    dw5x5_kernel<<<dim3(16 * DIMC), 256, 0, stream>>>(
        v_map, w_lepe, b_lepe, lepe);

    // 3) fused windowed attention (one block per window x head)
    lepe_attn_kernel<<<dim3(NWIN, HEADS), 256, 0, stream>>>(
        x, q_map, v_map, lepe, attn_input, temperature, out_img, out_attn);
}